// SimpleBatchedPKEModel_20727512170880
// MI455X (gfx1250) — compile-verified
//
#include <hip/hip_runtime.h>
#include <hip/hip_bf16.h>

#define BB 16
#define NN 512
#define DD 12
#define HID 32
#define HEADS 4
#define CC 8

typedef __attribute__((ext_vector_type(2))) float v2f;
typedef __attribute__((ext_vector_type(8))) float v8f;

// D = A(16x4, f32) * B(4x16, f32) + C(16x16, f32)
// 8-arg form: (neg_a, A, neg_b, B, c_mod, C, reuse_a, reuse_b)
__device__ __forceinline__ v8f wmma_f32_4(v2f a, v2f b, v8f c) {
  return __builtin_amdgcn_wmma_f32_16x16x4_f32(false, a, false, b, (short)0, c,
                                               false, false);
}

// ---------------------------------------------------------------------------
// k1: h = relu(x @ Wp^T + bp);  g = h @ Wg^T;  a_s/a_d = head dots with g.
// One wave per 16-row tile (512 tiles). WMMA f32 16x16x4, K chained.
// A-frag layout (ISA 7.12.2): lanes 0-15 hold M=lane, VGPR0/1 = K0/K1;
// lanes 16-31 hold M=lane-16, VGPR0/1 = K2/K3. B-frag mirrored (N across
// lanelo, K split across halves/regs).
// ---------------------------------------------------------------------------
__global__ __launch_bounds__(256) void k1_proj(
    const float* __restrict__ x, const float* __restrict__ Wp,
    const float* __restrict__ bp, const float* __restrict__ Wg,
    const float* __restrict__ att_src, const float* __restrict__ att_dst,
    float* __restrict__ g_out, float* __restrict__ as_out,
    float* __restrict__ ad_out) {
  __shared__ float lds[8 * 16 * 32];
  const int tid = threadIdx.x;
  const int w = tid >> 5;
  const int lane = tid & 31;
  const int half = lane >> 4;
  const int lanelo = lane & 15;
  const int t = blockIdx.x * 8 + w;  // tile id 0..511
  const int b = t >> 5;              // 32 tiles per batch
  const int n0 = (t & 31) << 4;
  float* hsh = &lds[w * 512];

  // ---- h tile: 16x32, K=12 (3 WMMA steps per 16-col tile) ----
  v8f h0 = {}, h1 = {};
  const float* xrow = x + (size_t)(b * NN + n0 + lanelo) * DD;
#pragma unroll
  for (int kk = 0; kk < 3; ++kk) {
    const int k = kk * 4 + 2 * half;
    v2f a;   a.x = xrow[k];                    a.y = xrow[k + 1];
    v2f b0;  b0.x = Wp[lanelo * DD + k];       b0.y = Wp[lanelo * DD + k + 1];
    v2f b1v; b1v.x = Wp[(16 + lanelo) * DD + k];
             b1v.y = Wp[(16 + lanelo) * DD + k + 1];
    h0 = wmma_f32_4(a, b0, h0);
    h1 = wmma_f32_4(a, b1v, h1);
  }
  const float bp0 = bp[lanelo], bp1 = bp[16 + lanelo];
#pragma unroll
  for (int r = 0; r < 8; ++r) {
    const int row = r + 8 * half;
    hsh[row * 32 + lanelo]      = fmaxf(h0[r] + bp0, 0.f);
    hsh[row * 32 + 16 + lanelo] = fmaxf(h1[r] + bp1, 0.f);
  }
  __syncthreads();

  // ---- g tile: 16x32, K=32 (8 WMMA steps per 16-col tile) ----
  v8f g0 = {}, g1 = {};
#pragma unroll
  for (int kk = 0; kk < 8; ++kk) {
    const int k = kk * 4 + 2 * half;
    v2f a;   a.x = hsh[lanelo * 32 + k];       a.y = hsh[lanelo * 32 + k + 1];
    v2f b0;  b0.x = Wg[lanelo * HID + k];      b0.y = Wg[lanelo * HID + k + 1];
    v2f b1v; b1v.x = Wg[(16 + lanelo) * HID + k];
             b1v.y = Wg[(16 + lanelo) * HID + k + 1];
    g0 = wmma_f32_4(a, b0, g0);
    g1 = wmma_f32_4(a, b1v, g1);
  }
  float* gg = g_out + (size_t)(b * NN + n0) * HID;
#pragma unroll
  for (int r = 0; r < 8; ++r) {
    const int row = r + 8 * half;
    gg[row * 32 + lanelo]      = g0[r];
    gg[row * 32 + 16 + lanelo] = g1[r];
  }
  __syncthreads();
#pragma unroll
  for (int r = 0; r < 8; ++r) {
    const int row = r + 8 * half;
    hsh[row * 32 + lanelo]      = g0[r];
    hsh[row * 32 + 16 + lanelo] = g1[r];
  }
  __syncthreads();

  // ---- a_s / a_d: 16 rows x 4 heads, each lane does 2 (row,head) pairs ----
  for (int p = lane; p < 64; p += 32) {
    const int m = p >> 2, hh = p & 3;
    float as = 0.f, ad = 0.f;
#pragma unroll
    for (int c = 0; c < CC; ++c) {
      const float gv = hsh[m * 32 + hh * CC + c];
      as += gv * att_src[hh * CC + c];
      ad += gv * att_dst[hh * CC + c];
    }
    as_out[(size_t)(b * NN + n0 + m) * HEADS + hh] = as;
    ad_out[(size_t)(b * NN + n0 + m) * HEADS + hh] = ad;
  }
}

// ---------------------------------------------------------------------------
// k2: masked softmax over i (two-pass, fused) + weighted aggregation of g.
// One thread per (b, j, h). adj reads are j-contiguous -> coalesced;
// a_s / g reads are wave-uniform -> broadcast.
// ---------------------------------------------------------------------------
__global__ __launch_bounds__(128) void k2_attn(
    const int* __restrict__ adj, const float* __restrict__ g,
    const float* __restrict__ a_s, const float* __restrict__ a_d,
    const float* __restrict__ bias_g, float* __restrict__ hgnn) {
  const int tid = threadIdx.x;
  const int b = blockIdx.x >> 4;                         // 16 j-tiles per batch
  const int j = ((blockIdx.x & 15) << 5) + (tid & 31);
  const int h = tid >> 5;
  const float ad = a_d[(size_t)(b * NN + j) * HEADS + h];
  const int* adjb = adj + (size_t)b * NN * NN;
  const float* asb = a_s + (size_t)b * NN * HEADS + h;

  float m = -3.0e38f;
  for (int i = 0; i < NN; ++i) {
    float l = asb[i * HEADS] + ad;
    l = l > 0.f ? l : 0.2f * l;  // leaky_relu(0.2)
    const bool on = (adjb[(size_t)i * NN + j] != 0) | (i == j);
    l = on ? l : -1.0e9f;
    m = fmaxf(m, l);
  }
  float s = 0.f;
  float acc[CC] = {0.f, 0.f, 0.f, 0.f, 0.f, 0.f, 0.f, 0.f};
  const float* gb = g + (size_t)b * NN * HID + h * CC;
  for (int i = 0; i < NN; ++i) {
    float l = asb[i * HEADS] + ad;
    l = l > 0.f ? l : 0.2f * l;
    const bool on = (adjb[(size_t)i * NN + j] != 0) | (i == j);
    l = on ? l : -1.0e9f;
    const float p = __expf(l - m);
    s += p;
    const float4 gv0 = *(const float4*)(gb + (size_t)i * HID);
    const float4 gv1 = *(const float4*)(gb + (size_t)i * HID + 4);
    acc[0] += p * gv0.x; acc[1] += p * gv0.y;
    acc[2] += p * gv0.z; acc[3] += p * gv0.w;
    acc[4] += p * gv1.x; acc[5] += p * gv1.y;
    acc[6] += p * gv1.z; acc[7] += p * gv1.w;
  }
  const float inv = 1.f / s;
  float* o = hgnn + (size_t)(b * NN + j) * HID + h * CC;
#pragma unroll
  for (int c = 0; c < CC; ++c) o[c] = acc[c] * inv + bias_g[h * CC + c];
}

// ---------------------------------------------------------------------------
// k3: ti = h_gnn @ W1[:, :32]^T,  tj = h_gnn @ W1[:, 32:]^T  via WMMA, K=32.
// One wave per 16-row tile; 4 accumulator tiles per wave.
// ---------------------------------------------------------------------------
__global__ __launch_bounds__(256) void k3_edge_proj(
    const float* __restrict__ hgnn, const float* __restrict__ W1,
    float* __restrict__ ti, float* __restrict__ tj) {
  const int tid = threadIdx.x;
  const int lane = tid & 31;
  const int half = lane >> 4;
  const int lanelo = lane & 15;
  const int t = blockIdx.x * 8 + (tid >> 5);  // 0..511
  const int n0 = t << 4;                      // global row (b folded in)
  v8f ti0 = {}, ti1 = {}, tj0 = {}, tj1 = {};
  const float* arow = hgnn + (size_t)(n0 + lanelo) * HID;
#pragma unroll
  for (int kk = 0; kk < 8; ++kk) {
    const int k = kk * 4 + 2 * half;
    v2f a; a.x = arow[k]; a.y = arow[k + 1];
    v2f bi0; bi0.x = W1[lanelo * 64 + k];            bi0.y = W1[lanelo * 64 + k + 1];
    v2f bi1; bi1.x = W1[(16 + lanelo) * 64 + k];     bi1.y = W1[(16 + lanelo) * 64 + k + 1];
    v2f bj0; bj0.x = W1[lanelo * 64 + 32 + k];       bj0.y = W1[lanelo * 64 + 33 + k];
    v2f bj1; bj1.x = W1[(16 + lanelo) * 64 + 32 + k];
             bj1.y = W1[(16 + lanelo) * 64 + 33 + k];
    ti0 = wmma_f32_4(a, bi0, ti0);
    ti1 = wmma_f32_4(a, bi1, ti1);
    tj0 = wmma_f32_4(a, bj0, tj0);
    tj1 = wmma_f32_4(a, bj1, tj1);
  }
#pragma unroll
  for (int r = 0; r < 8; ++r) {
    const size_t row = (size_t)(n0 + r + 8 * half) * HID;
    ti[row + lanelo]      = ti0[r];
    ti[row + 16 + lanelo] = ti1[r];
    tj[row + lanelo]      = tj0[r];
    tj[row + 16 + lanelo] = tj1[r];
  }
}

// ---------------------------------------------------------------------------
// k4: out[b,i,j] = (adj && i!=j) ? sigmoid(sum_h relu(ti[i,h]+tj[j,h]+b1)*w2+b2)
//                                : 0.   Block = 32 i x 256 j tile. ti tile in
// LDS (wave-uniform reads), tj row (+b1) in 32 regs. Coalesced adj/out.
// ---------------------------------------------------------------------------
__global__ __launch_bounds__(256) void k4_score(
    const int* __restrict__ adj, const float* __restrict__ ti,
    const float* __restrict__ tj, const float* __restrict__ b1,
    const float* __restrict__ w2, const float* __restrict__ b2,
    float* __restrict__ out) {
  __shared__ float tish[32 * 32];
  __shared__ float w2sh[32];
  const int tid = threadIdx.x;
  const int blk = blockIdx.x;
  const int b = blk >> 5;            // 32 blocks per batch (16 i-tiles * 2 j-tiles)
  const int i0 = ((blk >> 1) & 15) << 5;
  const int j = ((blk & 1) << 8) + tid;

  for (int idx = tid; idx < 1024; idx += 256)
    tish[idx] = ti[(size_t)(b * NN + i0 + (idx >> 5)) * HID + (idx & 31)];
  if (tid < 32) w2sh[tid] = w2[tid];

  float tjreg[32];
  {
    const float4* tp = (const float4*)(tj + (size_t)(b * NN + j) * HID);
#pragma unroll
    for (int q = 0; q < 8; ++q) {
      const float4 v = tp[q];
      tjreg[4 * q + 0] = v.x + b1[4 * q + 0];
      tjreg[4 * q + 1] = v.y + b1[4 * q + 1];
      tjreg[4 * q + 2] = v.z + b1[4 * q + 2];
      tjreg[4 * q + 3] = v.w + b1[4 * q + 3];
    }
  }
  const float b2v = b2[0];
  __syncthreads();

  const int* adjrow = adj + ((size_t)b * NN + i0) * NN + j;
  float* orow = out + ((size_t)b * NN + i0) * NN + j;
  for (int il = 0; il < 32; ++il) {
    const int i = i0 + il;
    const int e = adjrow[(size_t)il * NN];
    float r = 0.f;
    if ((e != 0) & (i != j)) {
      float acc = 0.f;
      const float* trow = &tish[il * 32];
#pragma unroll
      for (int hh = 0; hh < 32; ++hh)
        acc += fmaxf(trow[hh] + tjreg[hh], 0.f) * w2sh[hh];
      r = 1.f / (1.f + __expf(-(acc + b2v)));
    }
    orow[(size_t)il * NN] = r;
  }
}

extern "C" void kernel_launch(void* const* d_in, const int* in_sizes, int n_in,
                              void* d_out, int out_size, void* d_ws,
                              size_t ws_size, hipStream_t stream) {
  const float* x       = (const float*)d_in[0];
  const int*   adj     = (const int*)d_in[1];
  const float* Wp      = (const float*)d_in[2];
  const float* bp      = (const float*)d_in[3];
  const float* Wg      = (const float*)d_in[4];
  const float* att_src = (const float*)d_in[5];
  const float* att_dst = (const float*)d_in[6];
  const float* bias_g  = (const float*)d_in[7];
  const float* W1      = (const float*)d_in[8];
  const float* b1      = (const float*)d_in[9];
  const float* w2      = (const float*)d_in[10];
  const float* b2      = (const float*)d_in[11];
  float* out = (float*)d_out;

  // workspace layout (floats): g | a_s | a_d | h_gnn | ti | tj  (~4.3 MB)
  float* g    = (float*)d_ws;          // B*N*32 = 262144
  float* a_s  = g + BB * NN * HID;     // B*N*4
  float* a_d  = a_s + BB * NN * HEADS;
  float* hgnn = a_d + BB * NN * HEADS; // B*N*32
  float* ti   = hgnn + BB * NN * HID;  // B*N*32
  float* tjw  = ti + BB * NN * HID;    // B*N*32

  hipLaunchKernelGGL(k1_proj, dim3(64), dim3(256), 0, stream, x, Wp, bp, Wg,
                     att_src, att_dst, g, a_s, a_d);
  hipLaunchKernelGGL(k2_attn, dim3(256), dim3(128), 0, stream, adj, g, a_s,
                     a_d, bias_g, hgnn);
  hipLaunchKernelGGL(k3_edge_proj, dim3(64), dim3(256), 0, stream, hgnn, W1,
                     ti, tjw);
  hipLaunchKernelGGL(k4_score, dim3(512), dim3(256), 0, stream, adj, ti, tjw,
                     b1, w2, b2, out);
}